// MMD_Loss_20066087206949
// MI455X (gfx1250) — compile-verified
//
#include <hip/hip_runtime.h>
#include <math.h>

// MMD loss, algebraically reduced:
//   sum(L2) = 2n*sum(sq) - 2*||colsum||^2   (no GEMM, no 268MB matrix)
//   loss needs only 4*B shifted-diagonal dot products -> fp32 WMMA 16x16x4 tiles,
//   take the diagonal of each accumulated 16x16 P*Q^T tile.
// Memory-bound: ~25MB traffic ~= 1us at 23.3 TB/s. fp32 throughout (matches jnp f32).

typedef __attribute__((ext_vector_type(2))) float v2f;
typedef __attribute__((ext_vector_type(8))) float v8f;

#define Bn 4096
#define Dn 256
// ws layout (floats)
#define WS_SQ   0                  // 8192 row sq-norms (src then tgt)
#define WS_COL  8192               // 64*256 partial column sums
#define WS_BW   (WS_COL + 64*256)  // 1 bandwidth
#define WS_PART (WS_BW + 1)        // 256 per-block loss partials

__global__ __launch_bounds__(256)
void k_rowsq(const float* __restrict__ src, const float* __restrict__ tgt,
             float* __restrict__ ws) {
  int wave = threadIdx.x >> 5, lane = threadIdx.x & 31;
  int row = blockIdx.x * 8 + wave;               // 1024 blocks * 8 waves = 8192 rows
  const float* p = (row < Bn) ? (src + (size_t)row * Dn)
                              : (tgt + (size_t)(row - Bn) * Dn);
  float acc = 0.f;
  #pragma unroll
  for (int d = lane; d < Dn; d += 32) { float v = p[d]; acc += v * v; }
  #pragma unroll
  for (int off = 16; off > 0; off >>= 1) acc += __shfl_xor(acc, off, 32);
  if (lane == 0) ws[WS_SQ + row] = acc;
}

__global__ __launch_bounds__(256)
void k_colsum(const float* __restrict__ src, const float* __restrict__ tgt,
              float* __restrict__ ws) {
  int t = threadIdx.x;
  int r0 = blockIdx.x * 128;                     // 64 blocks * 128 rows = 8192
  float acc = 0.f;
  for (int rr = 0; rr < 128; ++rr) {
    int r = r0 + rr;
    acc += (r < Bn) ? src[(size_t)r * Dn + t]
                    : tgt[(size_t)(r - Bn) * Dn + t];
  }
  ws[WS_COL + blockIdx.x * 256 + t] = acc;
}

__global__ __launch_bounds__(256)
void k_bw(float* __restrict__ ws) {
  __shared__ float redS[256], redN[256];
  int t = threadIdx.x;
  float s = 0.f;
  for (int it = 0; it < 32; ++it) s += ws[WS_SQ + t + it * 256];
  float col = 0.f;
  for (int b = 0; b < 64; ++b) col += ws[WS_COL + b * 256 + t];
  redS[t] = s; redN[t] = col * col;
  __syncthreads();
  for (int off = 128; off > 0; off >>= 1) {
    if (t < off) { redS[t] += redS[t + off]; redN[t] += redN[t + off]; }
    __syncthreads();
  }
  if (t == 0) {
    float n = 2.0f * Bn;
    float sumL2 = 2.0f * n * redS[0] - 2.0f * redN[0];
    float bw = sumL2 / (n * n - n);
    bw *= 0.25f;                                  // /= KERNEL_MUL^(KERNEL_NUM//2)
    ws[WS_BW] = bw;
  }
}

__device__ __forceinline__ float pick8(v8f c, int k) {
  float r = c[0];
  if (k == 1) r = c[1]; if (k == 2) r = c[2]; if (k == 3) r = c[3];
  if (k == 4) r = c[4]; if (k == 5) r = c[5]; if (k == 6) r = c[6];
  if (k == 7) r = c[7];
  return r;
}

__device__ __forceinline__ float gauss5(float t) {  // t = -L2/bw
  return expf(t) + expf(t * 0.5f) + expf(t * 0.25f) +
         expf(t * 0.125f) + expf(t * 0.0625f);
}

// One wave per 16-row group: 4 accumulated 16x16 fp32 WMMA tiles, diagonals only.
__global__ __launch_bounds__(32)
void k_pairs(const float* __restrict__ src, const float* __restrict__ tgt,
             float* __restrict__ ws) {
  int lane = threadIdx.x;
  int i0 = blockIdx.x * 16;                      // 256 blocks cover B=4096 rows
  int m16  = (lane < 16) ? lane : lane - 16;     // matrix row this lane feeds
  int kofs = (lane < 16) ? 0 : 2;                // K sub-offset per fp32 A/B layout
  int rowA = i0 + m16;
  int rowB = (rowA + 1) & (Bn - 1);              // shifted diagonal (i+1)%B
  const float* pSi = src + (size_t)rowA * Dn + kofs;
  const float* pSj = src + (size_t)rowB * Dn + kofs;
  const float* pTi = tgt + (size_t)rowA * Dn + kofs;
  const float* pTj = tgt + (size_t)rowB * Dn + kofs;

  v8f ss = {}, tt = {}, st = {}, ts = {};
  #pragma unroll 4
  for (int kk = 0; kk < Dn; kk += 4) {
    v2f aSi = *(const v2f*)(pSi + kk);
    v2f aSj = *(const v2f*)(pSj + kk);
    v2f aTi = *(const v2f*)(pTi + kk);
    v2f aTj = *(const v2f*)(pTj + kk);
    // D[m][n] += sum_k P[m][kk+k] * Q[n][kk+k]  (frag works as either A or B)
    ss = __builtin_amdgcn_wmma_f32_16x16x4_f32(false, aSi, false, aSj, (short)0, ss, false, false);
    tt = __builtin_amdgcn_wmma_f32_16x16x4_f32(false, aTi, false, aTj, (short)0, tt, false, false);
    st = __builtin_amdgcn_wmma_f32_16x16x4_f32(false, aSi, false, aTj, (short)0, st, false, false);
    ts = __builtin_amdgcn_wmma_f32_16x16x4_f32(false, aSj, false, aTi, (short)0, ts, false, false);
  }

  // Diagonal of 16x16 f32 C/D: m=0..7 in lane m comp m; m=8..15 in lane m+16 comp m-8.
  bool active = (lane < 8) || (lane >= 24);
  int m = (lane < 8) ? lane : lane - 16;
  int comp = m & 7;
  float dss = pick8(ss, comp), dtt = pick8(tt, comp);
  float dst = pick8(st, comp), dts = pick8(ts, comp);

  float val = 0.f;
  if (active) {
    int i = i0 + m, j = (i + 1) & (Bn - 1);
    float sqsi = ws[WS_SQ + i],      sqsj = ws[WS_SQ + j];
    float sqti = ws[WS_SQ + Bn + i], sqtj = ws[WS_SQ + Bn + j];
    float ninv = -1.0f / ws[WS_BW];
    float La = sqsi + sqsj - 2.f * dss;   // K[i, j]
    float Lb = sqti + sqtj - 2.f * dtt;   // K[B+i, B+j]
    float Lc = sqsi + sqtj - 2.f * dst;   // K[i, B+j]
    float Ld = sqsj + sqti - 2.f * dts;   // K[j, B+i]
    val = gauss5(La * ninv) + gauss5(Lb * ninv)
        - gauss5(Lc * ninv) - gauss5(Ld * ninv);
  }
  #pragma unroll
  for (int off = 16; off > 0; off >>= 1) val += __shfl_xor(val, off, 32);
  if (lane == 0) ws[WS_PART + blockIdx.x] = val;
}

__global__ __launch_bounds__(256)
void k_final(const float* __restrict__ ws, float* __restrict__ out) {
  __shared__ float red[256];
  int t = threadIdx.x;
  red[t] = ws[WS_PART + t];
  __syncthreads();
  for (int off = 128; off > 0; off >>= 1) {
    if (t < off) red[t] += red[t + off];
    __syncthreads();
  }
  if (t == 0) out[0] = red[0] * (1.0f / (float)Bn);
}

extern "C" void kernel_launch(void* const* d_in, const int* in_sizes, int n_in,
                              void* d_out, int out_size, void* d_ws, size_t ws_size,
                              hipStream_t stream) {
  const float* src = (const float*)d_in[0];
  const float* tgt = (const float*)d_in[1];
  float* ws = (float*)d_ws;
  float* out = (float*)d_out;
  k_rowsq <<<1024, 256, 0, stream>>>(src, tgt, ws);
  k_colsum<<<  64, 256, 0, stream>>>(src, tgt, ws);
  k_bw    <<<   1, 256, 0, stream>>>(ws);
  k_pairs <<< 256,  32, 0, stream>>>(src, tgt, ws);
  k_final <<<   1, 256, 0, stream>>>(ws, out);
}